// CentroidAttention_70403103916528
// MI455X (gfx1250) — compile-verified
//
#include <hip/hip_runtime.h>
#include <hip/hip_bf16.h>

typedef __attribute__((ext_vector_type(2))) float v2f;
typedef __attribute__((ext_vector_type(8))) float v8f;

#define TT 16
#define DD 64

// One wave (32 lanes) per batch element n. Exact f32 compute via chained
// v_wmma_f32_16x16x4_f32. Requires full EXEC at every WMMA: kernel is only
// launched for the 8-wave-divisible bulk of N, with no divergence before the
// last WMMA except the reconverging predicated stores.
__global__ __launch_bounds__(256) void centroid_attn_wmma(
    const float* __restrict__ c_ft,   // [N, 16, 64]
    const float* __restrict__ ft,     // [N, 64]
    float* __restrict__ out)          // [N, 64]
{
  const int lane = threadIdx.x & 31;
  const int wave = threadIdx.x >> 5;
  const int n    = blockIdx.x * 8 + wave;

  const int q = lane & 15;   // column / row-in-half index
  const int g = lane >> 4;   // which half of the wave

  const float* Cn = c_ft + (size_t)n * (TT * DD);
  const float* Fn = ft   + (size_t)n * DD;

  // ---- Phase 1: att[j] = sum_d C[j,d] * f[d], j=0..15 -------------------
  // A[m,k] = f[4i+k] (rows replicated)  -> lane holds {f[4i+2g], f[4i+2g+1]}
  // B[k,j] = C[t=j, 4i+k]               -> lane holds {C[q,4i+2g], C[q,4i+2g+1]}
  // Accumulate over i=0..15 => D[m,j] = att[j] replicated across rows.
  v8f acc = {};
  #pragma unroll
  for (int i = 0; i < 16; ++i) {
    const int k0 = 4 * i + 2 * g;                 // even -> 8B aligned loads
    v2f a = *(const v2f*)(Fn + k0);
    v2f b = *(const v2f*)(Cn + q * DD + k0);
    acc = __builtin_amdgcn_wmma_f32_16x16x4_f32(
        false, a, false, b, (short)0, acc, false, false);
  }
  float att = acc[0];                             // att[q] in lanes q and q+16

  // ---- softmax over the 16 att values (xor-reduce within 16-lane halves) --
  float m = att;
  #pragma unroll
  for (int off = 1; off < 16; off <<= 1)
    m = fmaxf(m, __shfl_xor(m, off, 32));
  float e = __expf(att - m);
  float s = e;
  #pragma unroll
  for (int off = 1; off < 16; off <<= 1)
    s += __shfl_xor(s, off, 32);
  const float w = e / s;                          // w[q]

  // ---- Phase 2: out[d] = sum_t w[t] * C[t,d] ----------------------------
  // 4 d-tiles (dt) x 4 t-chunks (i):
  //   A[m,k] = w[4i+k] replicated; B[k,j] = C[t=4i+k, 16*dt + j]
  // B loads are fully coalesced 64B row segments.
  #pragma unroll
  for (int dt = 0; dt < 4; ++dt) {
    v8f acc2 = {};
    #pragma unroll
    for (int i = 0; i < 4; ++i) {
      const int t0 = 4 * i + 2 * g;
      v2f a2;
      a2.x = __shfl(w, 4 * i + 2 * g,     32);
      a2.y = __shfl(w, 4 * i + 2 * g + 1, 32);
      v2f b2;
      b2.x = Cn[(t0 + 0) * DD + 16 * dt + q];
      b2.y = Cn[(t0 + 1) * DD + 16 * dt + q];
      acc2 = __builtin_amdgcn_wmma_f32_16x16x4_f32(
          false, a2, false, b2, (short)0, acc2, false, false);
    }
    if (g == 0)                                   // reconverges before next WMMA
      out[(size_t)n * DD + 16 * dt + q] = acc2[0];
  }
}

// Pure-VALU tail for any n not covered by the 8-wave blocks (launches zero
// blocks for N=200000). One thread per output element; EXEC-guarded, no WMMA.
__global__ __launch_bounds__(256) void centroid_attn_tail(
    const float* __restrict__ c_ft, const float* __restrict__ ft,
    float* __restrict__ out, int n0, int N)
{
  const int idx = blockIdx.x * blockDim.x + threadIdx.x;
  const int n = n0 + idx / DD;
  const int d = idx % DD;
  if (n >= N) return;
  const float* Cn = c_ft + (size_t)n * (TT * DD);
  const float* Fn = ft   + (size_t)n * DD;
  float att[TT];
  float mx = -3.4e38f;
  for (int t = 0; t < TT; ++t) {
    float a = 0.f;
    for (int k = 0; k < DD; ++k) a += Cn[t * DD + k] * Fn[k];
    att[t] = a;
    mx = fmaxf(mx, a);
  }
  float s = 0.f;
  for (int t = 0; t < TT; ++t) { att[t] = __expf(att[t] - mx); s += att[t]; }
  const float inv = 1.f / s;
  float o = 0.f;
  for (int t = 0; t < TT; ++t) o += att[t] * inv * Cn[t * DD + d];
  out[(size_t)n * DD + d] = o;
}

extern "C" void kernel_launch(void* const* d_in, const int* in_sizes, int n_in,
                              void* d_out, int out_size, void* d_ws, size_t ws_size,
                              hipStream_t stream) {
  const float* c_ft = (const float*)d_in[0];   // [N,16,64] f32
  const float* ft   = (const float*)d_in[1];   // [N,64]    f32
  float* out        = (float*)d_out;           // [N,64]    f32

  const int N      = in_sizes[1] / DD;
  const int nbulk  = N / 8;                    // 8 waves (one n each) per block
  const int n_done = nbulk * 8;

  if (nbulk > 0)
    centroid_attn_wmma<<<nbulk, 256, 0, stream>>>(c_ft, ft, out);

  const int rem = N - n_done;                  // 0 for N=200000
  if (rem > 0) {
    const int threads = rem * DD;
    centroid_attn_tail<<<(threads + 255) / 256, 256, 0, stream>>>(
        c_ft, ft, out, n_done, N);
  }
}